// MaxCombinator_33457795236047
// MI455X (gfx1250) — compile-verified
//
#include <hip/hip_runtime.h>

#define TPB 256
#define UNROLL 16

// ---------------------------------------------------------------------------
// Phase 1: each block streams a contiguous chunk of rows, keeping a full
// [S x 256] f32 max-accumulator in LDS (64 KB). Thread t owns column t of
// every segment -> no races, no barriers. Epilogue either stores the partial
// tile to workspace (two-phase mode) or atomic-maxes into d_out directly.
// ---------------------------------------------------------------------------
__global__ __launch_bounds__(TPB)
void seg_max_phase1(const float* __restrict__ emb,
                    const int*   __restrict__ sidx,
                    float*       __restrict__ dst,
                    int N, int S, int rowsPerBlock, int atomicMode)
{
    __shared__ float acc[64 * 256];   // 64 KB of the WGP's 320 KB LDS
    const int tid = threadIdx.x;
    const float NEG_INF = -__builtin_inff();

#pragma unroll
    for (int s = 0; s < 64; ++s)
        acc[s * 256 + tid] = NEG_INF;
    // No barrier needed: each (segment, col=tid) cell is private to thread tid.

    int rowBeg = blockIdx.x * rowsPerBlock;
    int rowEnd = rowBeg + rowsPerBlock;
    if (rowEnd > N) rowEnd = N;

    if (rowBeg < rowEnd) {
        const float* rp = emb + (size_t)rowBeg * 256 + tid;  // coalesced column stream
        const int*   ip = sidx + rowBeg;                     // block-uniform -> s_load
        const int n = rowEnd - rowBeg;
        int i = 0;
        for (; i + UNROLL <= n; i += UNROLL) {
            // Keep the scalar index stream ahead (global_prefetch_b8).
            // Clamp so non-speculative prefetch never walks past the array.
            {
                int pf = rowBeg + i + 2048;
                if (pf > N - 1) pf = N - 1;
                __builtin_prefetch(sidx + pf, 0, 0);
            }

            // UNROLL independent coalesced loads in flight per wave
            float v[UNROLL];
            int   sg[UNROLL];
#pragma unroll
            for (int k = 0; k < UNROLL; ++k)
                v[k] = rp[(size_t)(i + k) * 256];
#pragma unroll
            for (int k = 0; k < UNROLL; ++k)
                sg[k] = ip[i + k];
#pragma unroll
            for (int k = 0; k < UNROLL; ++k) {
                float* a = &acc[sg[k] * 256 + tid];
                *a = fmaxf(*a, v[k]);
            }
        }
        for (; i < n; ++i) {
            float v = rp[(size_t)i * 256];
            int   s = ip[i];
            float* a = &acc[s * 256 + tid];
            *a = fmaxf(*a, v);
        }
    }

    if (atomicMode) {
        for (int s = 0; s < S; ++s)
            atomicMax(&dst[(size_t)s * 256 + tid], acc[s * 256 + tid]);
    } else {
        float* o = dst + (size_t)blockIdx.x * (size_t)S * 256;
        for (int s = 0; s < S; ++s)
            o[(size_t)s * 256 + tid] = acc[s * 256 + tid];
    }
}

// ---------------------------------------------------------------------------
// Phase 2: reduce P per-block partial tiles (each S*256 floats) down to d_out.
// Re-reads P * 64 KB; trivially fast at 23.3 TB/s.
// ---------------------------------------------------------------------------
__global__ __launch_bounds__(TPB)
void seg_max_phase2(const float* __restrict__ part,
                    float*       __restrict__ out,
                    int total, int P)
{
    int e = blockIdx.x * TPB + threadIdx.x;
    if (e >= total) return;
    float m = -__builtin_inff();
#pragma unroll 4
    for (int p = 0; p < P; ++p)
        m = fmaxf(m, part[(size_t)p * total + e]);
    out[e] = m;
}

// d_out is poisoned before timing -> must (re)initialize to the max identity
// on paths that combine via atomics.
__global__ __launch_bounds__(TPB)
void seg_max_init(float* __restrict__ out, int n)
{
    int i = blockIdx.x * TPB + threadIdx.x;
    if (i < n) out[i] = -__builtin_inff();
}

// Generic fallback for unexpected shapes: one thread per element, atomic max.
__global__ __launch_bounds__(TPB)
void seg_max_generic(const float* __restrict__ emb,
                     const int*   __restrict__ sidx,
                     float*       __restrict__ out,
                     long long total, int D)
{
    long long i = (long long)blockIdx.x * TPB + threadIdx.x;
    if (i >= total) return;
    long long r = i / D;
    int       c = (int)(i - r * (long long)D);
    atomicMax(&out[(size_t)sidx[r] * D + c], emb[i]);
}

extern "C" void kernel_launch(void* const* d_in, const int* in_sizes, int n_in,
                              void* d_out, int out_size, void* d_ws, size_t ws_size,
                              hipStream_t stream)
{
    const float* emb  = (const float*)d_in[0];
    const int*   sidx = (const int*)d_in[1];
    float*       out  = (float*)d_out;

    const long long ND = in_sizes[0];
    const int       N  = in_sizes[1];
    const int       D  = (N > 0) ? (int)(ND / N) : 0;
    const int       S  = (D > 0) ? (out_size / D) : 0;

    if (D == 256 && S >= 1 && S <= 64 && N > 0) {
        const size_t partialBytes = (size_t)S * 256 * sizeof(float);
        int maxP = (ws_size >= partialBytes) ? (int)(ws_size / partialBytes) : 0;
        int P = 1024;                 // saturate ~all WGPs (5 x 64KB tiles fit per WGP)
        if (P > maxP) P = maxP;

        if (P >= 64) {
            // Two-phase: partials in workspace, then a tiny linear reduce over P.
            const int rpb = (N + P - 1) / P;
            seg_max_phase1<<<P, TPB, 0, stream>>>(emb, sidx, (float*)d_ws,
                                                  N, S, rpb, /*atomicMode=*/0);
            const int total = S * 256;
            seg_max_phase2<<<(total + TPB - 1) / TPB, TPB, 0, stream>>>(
                (const float*)d_ws, out, total, P);
        } else {
            // Workspace too small: init to -inf, combine with float atomic max.
            int P2 = 512;
            if (P2 > N) P2 = N;
            const int rpb = (N + P2 - 1) / P2;
            seg_max_init<<<(out_size + TPB - 1) / TPB, TPB, 0, stream>>>(out, out_size);
            seg_max_phase1<<<P2, TPB, 0, stream>>>(emb, sidx, out,
                                                   N, S, rpb, /*atomicMode=*/1);
        }
    } else {
        // Unexpected shape: fully generic atomic path.
        seg_max_init<<<(out_size + TPB - 1) / TPB, TPB, 0, stream>>>(out, out_size);
        const long long total  = ND;
        const long long blocks = (total + TPB - 1) / TPB;
        seg_max_generic<<<(unsigned)blocks, TPB, 0, stream>>>(emb, sidx, out, total, D);
    }
}